// _InitialPoseModel_31387620999481
// MI455X (gfx1250) — compile-verified
//
#include <hip/hip_runtime.h>
#include <stdint.h>

#define TPB 256
#define KCAND 10
#define MAXKEY 0xFFFFFFFFFFFFFFFFull

// Low 32 bits of a generic pointer to a __shared__ object are the LDS byte
// offset (flat->LDS mapping truncates to addr[31:0], ISA 10.2).
__device__ __forceinline__ unsigned lds_off_u32(const void* p) {
    return (unsigned)(uintptr_t)p;
}

__device__ __forceinline__ void async_gload_b32(unsigned lds_off, const void* g) {
    asm volatile("global_load_async_to_lds_b32 %0, %1, off"
                 :: "v"(lds_off), "v"((unsigned long long)(uintptr_t)g)
                 : "memory");
}
__device__ __forceinline__ void async_gload_b64(unsigned lds_off, const void* g) {
    asm volatile("global_load_async_to_lds_b64 %0, %1, off"
                 :: "v"(lds_off), "v"((unsigned long long)(uintptr_t)g)
                 : "memory");
}
__device__ __forceinline__ void wait_async_le4() {
    asm volatile("s_wait_asynccnt 0x4" ::: "memory");
}
__device__ __forceinline__ void wait_async_0() {
    asm volatile("s_wait_asynccnt 0x0" ::: "memory");
}

// One block per (batch b, keypoint-channel j). j in [0, nk]; j==nk selects cpt.
// Phase 1: double-buffered async global->LDS staging + per-thread register top-10.
// Phase 2: LDS merge, 10 rounds of wave32+LDS min-extraction (keys are unique).
// Phase 3: gather candidates, sigma-filtered mean per coordinate.
__global__ __launch_bounds__(TPB) void pose_vote_kernel(
    const float* __restrict__ pcld,      // [bs, n, 3]
    const float* __restrict__ kpts,      // [bs, n, nk, 3]
    const float* __restrict__ cpt,       // [bs, n, 1, 3]
    const float* __restrict__ seg,       // [bs, n, 2]
    float* __restrict__ out,             // [bs, nk+1, 3]
    int n, int nk)
{
    const int nc  = nk + 1;
    const int b   = blockIdx.x / nc;
    const int j   = blockIdx.x % nc;
    const int tid = threadIdx.x;

    const float* offbase;
    size_t stride;
    if (j < nk) { offbase = kpts + ((size_t)b * n * nk + (size_t)j) * 3; stride = (size_t)nk * 3; }
    else        { offbase = cpt  +  (size_t)b * n * 3;                   stride = 3; }
    const float* segb = seg + (size_t)b * n * 2;

    // Staging buffers: each thread consumes only its own slots -> no barriers,
    // only per-wave s_wait_asynccnt (async loads complete in order).
    __shared__ float offbuf[2][TPB * 3];   // 6 KB
    __shared__ float segbuf[2][TPB * 2];   // 4 KB
    __shared__ unsigned long long cand[TPB * KCAND];   // 20 KB
    __shared__ unsigned long long wmin[TPB / 32];
    __shared__ unsigned long long selk[KCAND];
    __shared__ float cans[KCAND * 3];

    const int ntiles = (n + TPB - 1) / TPB;

    auto issue_tile = [&](int t) {
        const int buf = t & 1;
        int p = t * TPB + tid;
        if (p >= n) p = n - 1;                        // clamp: keep EXEC uniform
        const float* o = offbase + (size_t)p * stride;
        unsigned lo = lds_off_u32(&offbuf[buf][tid * 3]);
        async_gload_b32(lo + 0, o + 0);
        async_gload_b32(lo + 4, o + 1);
        async_gload_b32(lo + 8, o + 2);
        async_gload_b64(lds_off_u32(&segbuf[buf][tid * 2]), segb + 2 * (size_t)p);
    };

    // ---- Phase 1: per-thread top-10 smallest keys (sorted ascending in regs) ----
    unsigned long long arr[KCAND];
    #pragma unroll
    for (int i = 0; i < KCAND; ++i) arr[i] = MAXKEY;

    issue_tile(0);
    for (int t = 0; t < ntiles; ++t) {
        if (t + 1 < ntiles) { issue_tile(t + 1); wait_async_le4(); }
        else               { wait_async_0(); }
        const int buf = t & 1;
        const int p = t * TPB + tid;
        if (p < n) {
            float s0 = segbuf[buf][tid * 2 + 0];
            float s1 = segbuf[buf][tid * 2 + 1];
            float infl = (s1 > s0) ? 1.0f : 1000.0f;   // argmax==0 (ties -> 0) => inflate
            float x = offbuf[buf][tid * 3 + 0] * infl;
            float y = offbuf[buf][tid * 3 + 1] * infl;
            float z = offbuf[buf][tid * 3 + 2] * infl;
            float nrm = sqrtf(x * x + y * y + z * z);
            unsigned long long key =
                ((unsigned long long)__float_as_uint(nrm) << 32) | (unsigned int)p;
            if (key < arr[KCAND - 1]) {
                unsigned long long cur = key;
                #pragma unroll
                for (int i = 0; i < KCAND; ++i) {      // branchless sorted insert
                    unsigned long long lo = cur < arr[i] ? cur : arr[i];
                    unsigned long long hi = cur < arr[i] ? arr[i] : cur;
                    arr[i] = lo; cur = hi;
                }
            }
        }
    }

    // ---- Phase 2: merge 256 sorted 10-lists via 10 min-extractions ----
    #pragma unroll
    for (int i = 0; i < KCAND; ++i) cand[tid * KCAND + i] = arr[i];
    __syncthreads();

    const int lane = tid & 31;
    const int wid  = tid >> 5;
    int ci = 0;                                        // cursor into my sorted list
    for (int r = 0; r < KCAND; ++r) {
        unsigned long long my = (ci < KCAND) ? cand[tid * KCAND + ci] : MAXKEY;
        unsigned long long v = my;
        #pragma unroll
        for (int off = 16; off > 0; off >>= 1) {       // wave32 min-reduce (2x u32 shuffles)
            unsigned int lo  = (unsigned int)v;
            unsigned int hi  = (unsigned int)(v >> 32);
            unsigned int olo = __shfl_down(lo, (unsigned)off, 32);
            unsigned int ohi = __shfl_down(hi, (unsigned)off, 32);
            unsigned long long o = ((unsigned long long)ohi << 32) | olo;
            v = v < o ? v : o;
        }
        if (lane == 0) wmin[wid] = v;
        __syncthreads();
        if (tid == 0) {
            unsigned long long g = wmin[0];
            #pragma unroll
            for (int w = 1; w < TPB / 32; ++w) g = g < wmin[w] ? g : wmin[w];
            selk[r] = g;
        }
        __syncthreads();
        if (ci < KCAND && my == selk[r]) ci++;         // keys unique -> exactly one advances
    }

    // ---- Phase 3: gather winners, inlier-filtered mean ----
    if (tid < KCAND) {
        int p = (int)(unsigned int)(selk[tid] & 0xFFFFFFFFull);
        const float* pc = pcld + ((size_t)b * n + (size_t)p) * 3;
        const float* o  = offbase + (size_t)p * stride;
        cans[tid * 3 + 0] = pc[0] + o[0];
        cans[tid * 3 + 1] = pc[1] + o[1];
        cans[tid * 3 + 2] = pc[2] + o[2];
    }
    __syncthreads();
    if (tid < 3) {
        const int c = tid;
        float m = 0.f;
        #pragma unroll
        for (int i = 0; i < KCAND; ++i) m += cans[i * 3 + c];
        m *= (1.0f / KCAND);
        float var = 0.f;
        #pragma unroll
        for (int i = 0; i < KCAND; ++i) { float d = cans[i * 3 + c] - m; var += d * d; }
        float sd = sqrtf(var * (1.0f / KCAND));
        float s = 0.f, cnt = 0.f;
        #pragma unroll
        for (int i = 0; i < KCAND; ++i) {
            float xv = cans[i * 3 + c];
            if (fabsf(xv - m) <= sd) { s += xv; cnt += 1.f; }
        }
        out[((size_t)b * (nk + 1) + (size_t)j) * 3 + c] = s / cnt;
    }
}

extern "C" void kernel_launch(void* const* d_in, const int* in_sizes, int n_in,
                              void* d_out, int out_size, void* d_ws, size_t ws_size,
                              hipStream_t stream) {
    (void)n_in; (void)d_ws; (void)ws_size;
    const float* pcld = (const float*)d_in[0];   // [bs, n, 3]
    const float* kpts = (const float*)d_in[1];   // [bs, n, nk, 3]
    const float* cpt  = (const float*)d_in[2];   // [bs, n, 1, 3]
    const float* seg  = (const float*)d_in[3];   // [bs, n, 2]
    float* out = (float*)d_out;

    // Derive shapes generically from flat sizes.
    const int nk = in_sizes[1] / in_sizes[0];            // kpts/pcld ratio = 8
    const int nc = nk + 1;                               // 9
    const int bs = out_size / (3 * nc);                  // 64
    const int n  = in_sizes[0] / (3 * bs);               // 12288

    dim3 grid((unsigned)(bs * nc));
    dim3 block(TPB);
    pose_vote_kernel<<<grid, block, 0, stream>>>(pcld, kpts, cpt, seg, out, n, nk);
}